// GCN_layers_24197845746329
// MI455X (gfx1250) — compile-verified
//
#include <hip/hip_runtime.h>
#include <hip/hip_bf16.h>

#define Nn 50000
#define Dd 128
#define Ee 800000
#define EP (Ee + Nn)          // edges + self loops
#define NEG_SLOPE 0.2f
#define LDS_STRIDE 132        // 128 + 4 pad: avoids 64-bank conflicts on column reads

typedef float v2f __attribute__((ext_vector_type(2)));
typedef float v8f __attribute__((ext_vector_type(8)));

// ---------- order-preserving float<->uint encoding for atomic max ----------
__device__ __forceinline__ unsigned fenc(float f) {
    unsigned u = __float_as_uint(f);
    return (u & 0x80000000u) ? ~u : (u | 0x80000000u);
}
__device__ __forceinline__ float fdec(unsigned e) {
    unsigned u = (e & 0x80000000u) ? (e & 0x7FFFFFFFu) : ~e;
    return __uint_as_float(u);
}

// =====================================================================
// Dual GEMM: xl = h @ Wl, xr = h @ Wr   (h: N x 128, W: 128 x 128)
// grid.x = N/16 = 3125 blocks, 256 threads = 8 waves.
// Wave w computes the 16x16 tile at columns [16w, 16w+16).
// Uses V_WMMA_F32_16X16X4_F32 (fp32-exact), K stepped by 4, 32 steps.
// =====================================================================
__global__ __launch_bounds__(256) void gemm_dual_wmma(
    const float* __restrict__ h,
    const float* __restrict__ Wl,
    const float* __restrict__ Wr,
    float* __restrict__ xl,
    float* __restrict__ xr)
{
    __shared__ float tile[16 * LDS_STRIDE];

    const int rowbase = blockIdx.x * 16;
    const int tid     = threadIdx.x;

    // ---- stage 16x128 tile of h into LDS: 512 float4's, 2 per thread ----
    #pragma unroll
    for (int j = 0; j < 2; ++j) {
        int idx  = tid + 256 * j;          // 0..511
        int row  = idx >> 5;               // 32 float4 per row
        int col  = (idx & 31) * 4;
        float4 v = *reinterpret_cast<const float4*>(h + (size_t)(rowbase + row) * Dd + col);
        *reinterpret_cast<float4*>(&tile[row * LDS_STRIDE + col]) = v;  // row*132+col is 16B aligned
    }
    __syncthreads();

    const int lane = tid & 31;
    const int wave = tid >> 5;
    const int col0 = wave * 16;
    const int m    = lane & 15;            // A-matrix row (and B/D column index)
    const int hi   = (lane >> 4) * 2;      // K sub-offset: lanes 0-15 -> K{0,1}, 16-31 -> K{2,3}

    v8f accL = {0.f, 0.f, 0.f, 0.f, 0.f, 0.f, 0.f, 0.f};
    v8f accR = {0.f, 0.f, 0.f, 0.f, 0.f, 0.f, 0.f, 0.f};

    #pragma unroll 8
    for (int kk = 0; kk < 32; ++kk) {
        const int k0 = kk * 4;
        // A fragment (16x4): lane L<16 holds row L, K={k0,k0+1}; lane L+16 holds K={k0+2,k0+3}
        v2f a;
        a.x = tile[m * LDS_STRIDE + k0 + hi];
        a.y = tile[m * LDS_STRIDE + k0 + hi + 1];
        // B fragments (4x16): vgpr0 = rows k0 / k0+2, vgpr1 = rows k0+1 / k0+3, cols across lanes
        v2f bl, br;
        bl.x = Wl[(size_t)(k0 + hi)     * Dd + col0 + m];
        bl.y = Wl[(size_t)(k0 + hi + 1) * Dd + col0 + m];
        br.x = Wr[(size_t)(k0 + hi)     * Dd + col0 + m];
        br.y = Wr[(size_t)(k0 + hi + 1) * Dd + col0 + m];
        accL = __builtin_amdgcn_wmma_f32_16x16x4_f32(false, a, false, bl, (short)0, accL, false, false);
        accR = __builtin_amdgcn_wmma_f32_16x16x4_f32(false, a, false, br, (short)0, accR, false, false);
    }

    // D layout: vgpr r -> row r (lanes 0-15) / row r+8 (lanes 16-31), col = lane&15
    const int rsel = (lane >> 4) * 8;
    #pragma unroll
    for (int r = 0; r < 8; ++r) {
        const size_t row = (size_t)(rowbase + r + rsel);
        xl[row * Dd + col0 + m] = accL[r];
        xr[row * Dd + col0 + m] = accR[r];
    }
}

// =====================================================================
// Zero accumulators: acc (N*D floats), menc (N uints), z (N floats)
// =====================================================================
__global__ __launch_bounds__(256) void zero_kernel(
    float* __restrict__ acc, unsigned* __restrict__ menc, float* __restrict__ z)
{
    const int i = blockIdx.x * blockDim.x + threadIdx.x;
    if (i < Nn * Dd) acc[i] = 0.0f;
    if (i < Nn) { menc[i] = 0u; z[i] = 0.0f; }
}

// =====================================================================
// Edge pass 1: score[e] = att . leaky_relu(xl[src] + xr[dst]);
//              atomicMax(menc[dst], enc(score)).  One wave per edge,
//              lane owns a float4 slice of D=128.
// =====================================================================
__global__ __launch_bounds__(256) void edge_score_kernel(
    const float* __restrict__ xl, const float* __restrict__ xr,
    const long long* __restrict__ ei, const float* __restrict__ att,
    float* __restrict__ score, unsigned* __restrict__ menc)
{
    const int widx = (int)((blockIdx.x * (unsigned)blockDim.x + threadIdx.x) >> 5);
    const int lane = threadIdx.x & 31;
    if (widx >= EP) return;

    long long s, d;
    if (widx < Ee) { s = ei[widx]; d = ei[Ee + widx]; }
    else           { s = widx - Ee; d = s; }

    const int d0 = lane * 4;
    const float4 lv = *reinterpret_cast<const float4*>(xl + (size_t)s * Dd + d0);
    const float4 rv = *reinterpret_cast<const float4*>(xr + (size_t)d * Dd + d0);
    const float4 av = *reinterpret_cast<const float4*>(att + d0);

    float t, acc = 0.0f;
    t = lv.x + rv.x; t = (t > 0.f) ? t : t * NEG_SLOPE; acc += av.x * t;
    t = lv.y + rv.y; t = (t > 0.f) ? t : t * NEG_SLOPE; acc += av.y * t;
    t = lv.z + rv.z; t = (t > 0.f) ? t : t * NEG_SLOPE; acc += av.z * t;
    t = lv.w + rv.w; t = (t > 0.f) ? t : t * NEG_SLOPE; acc += av.w * t;

    #pragma unroll
    for (int off = 16; off > 0; off >>= 1) acc += __shfl_xor(acc, off, 32);

    if (lane == 0) {
        score[widx] = acc;
        atomicMax(menc + d, fenc(acc));
    }
}

// =====================================================================
// Edge pass 2: w = exp(score - m[dst]); score[e] = w; z[dst] += w
// One thread per edge.
// =====================================================================
__global__ __launch_bounds__(256) void edge_exp_kernel(
    const long long* __restrict__ ei, float* __restrict__ score,
    const unsigned* __restrict__ menc, float* __restrict__ z)
{
    const int e = blockIdx.x * blockDim.x + threadIdx.x;
    if (e >= EP) return;
    const long long d = (e < Ee) ? ei[Ee + e] : (long long)(e - Ee);
    const float w = __expf(score[e] - fdec(menc[d]));
    score[e] = w;
    unsafeAtomicAdd(z + d, w);
}

// =====================================================================
// Edge pass 3: acc[dst] += (w / z[dst]) * xl[src].  Wave per edge,
// lane owns a float4 slice -> 4 hardware f32 atomic adds per lane.
// =====================================================================
__global__ __launch_bounds__(256) void edge_aggr_kernel(
    const float* __restrict__ xl, const long long* __restrict__ ei,
    const float* __restrict__ score, const float* __restrict__ z,
    float* __restrict__ acc)
{
    const int widx = (int)((blockIdx.x * (unsigned)blockDim.x + threadIdx.x) >> 5);
    const int lane = threadIdx.x & 31;
    if (widx >= EP) return;

    long long s, d;
    if (widx < Ee) { s = ei[widx]; d = ei[Ee + widx]; }
    else           { s = widx - Ee; d = s; }

    const float alpha = score[widx] / z[d];
    const int d0 = lane * 4;
    const float4 v = *reinterpret_cast<const float4*>(xl + (size_t)s * Dd + d0);
    float* o = acc + (size_t)d * Dd + d0;
    unsafeAtomicAdd(o + 0, alpha * v.x);
    unsafeAtomicAdd(o + 1, alpha * v.y);
    unsafeAtomicAdd(o + 2, alpha * v.z);
    unsafeAtomicAdd(o + 3, alpha * v.w);
}

// =====================================================================
// Epilogue: acc[i] = (acc[i] + b[col])  [+ ReLU for layers 0,1], in place
// =====================================================================
__global__ __launch_bounds__(256) void finalize_kernel(
    float* __restrict__ acc, const float* __restrict__ b, int do_relu)
{
    const int i = blockIdx.x * blockDim.x + threadIdx.x;
    if (i >= Nn * Dd) return;
    float v = acc[i] + b[i & (Dd - 1)];
    if (do_relu) v = fmaxf(v, 0.0f);
    acc[i] = v;
}

// =====================================================================
extern "C" void kernel_launch(void* const* d_in, const int* in_sizes, int n_in,
                              void* d_out, int out_size, void* d_ws, size_t ws_size,
                              hipStream_t stream) {
    const float*     x  = (const float*)d_in[0];
    const long long* ei = (const long long*)d_in[1];   // int64 [2, E]
    const float* Wls[3]  = {(const float*)d_in[2], (const float*)d_in[6],  (const float*)d_in[10]};
    const float* Wrs[3]  = {(const float*)d_in[3], (const float*)d_in[7],  (const float*)d_in[11]};
    const float* atts[3] = {(const float*)d_in[4], (const float*)d_in[8],  (const float*)d_in[12]};
    const float* bs[3]   = {(const float*)d_in[5], (const float*)d_in[9],  (const float*)d_in[13]};

    const size_t NB = (size_t)Nn * Dd * sizeof(float);   // 25.6 MB
    char* p = (char*)d_ws;
    float*    xl    = (float*)p;    p += NB;
    float*    xr    = (float*)p;    p += NB;
    float*    hA    = (float*)p;    p += NB;
    float*    hB    = (float*)p;    p += NB;
    float*    score = (float*)p;    p += (size_t)EP * sizeof(float);
    unsigned* menc  = (unsigned*)p; p += (size_t)Nn * sizeof(unsigned);
    float*    zbuf  = (float*)p;    p += (size_t)Nn * sizeof(float);
    (void)ws_size; (void)in_sizes; (void)n_in; (void)out_size;

    const int gGemm = Nn / 16;                    // 3125
    const int gNode = (Nn * Dd + 255) / 256;      // 25000
    const int gWave = (EP + 7) / 8;               // one wave per edge, 8 waves/block
    const int gEdge = (EP + 255) / 256;

    for (int l = 0; l < 3; ++l) {
        const float* hin = (l == 0) ? x : ((l == 1) ? hA : hB);
        float* acc = (l == 0) ? hA : ((l == 1) ? hB : (float*)d_out);

        gemm_dual_wmma<<<gGemm, 256, 0, stream>>>(hin, Wls[l], Wrs[l], xl, xr);
        zero_kernel<<<gNode, 256, 0, stream>>>(acc, menc, zbuf);
        edge_score_kernel<<<gWave, 256, 0, stream>>>(xl, xr, ei, atts[l], score, menc);
        edge_exp_kernel<<<gEdge, 256, 0, stream>>>(ei, score, menc, zbuf);
        edge_aggr_kernel<<<gWave, 256, 0, stream>>>(xl, ei, score, zbuf, acc);
        finalize_kernel<<<gNode, 256, 0, stream>>>(acc, bs[l], (l < 2) ? 1 : 0);
    }
}